// Sparsify2D_vol_987842478199
// MI455X (gfx1250) — compile-verified
//
#include <hip/hip_runtime.h>

// ---------------------------------------------------------------------------
// Per-row exact top-k (k = 50%) threshold via 4-round MSB radix select on the
// order-preserving float->uint key, then elementwise filter.
//   grid  = B (=128) workgroups, one row each
//   block = 1024 threads (32 wave32s -> fills one WGP)
// Data path: global -> LDS via async copies (ASYNCcnt), LDS banked histogram
// via ds_add_u32 atomics, non-temporal stores for the output.
// ---------------------------------------------------------------------------

#define THREADS   1024
#define ROW_ELEMS 196608              // 192*32*32
#define ROW_VEC   (ROW_ELEMS / 4)     // 49152 float4 per row
#define K_KEEP    (ROW_ELEMS / 2)     // 98304
#define NTILES    (ROW_VEC / THREADS) // 48 tiles of 1024 float4
#define NBANK     16

// Native clang vectors (HIP's float4 is a class type; builtins want these).
typedef float    v4f __attribute__((ext_vector_type(4)));
typedef unsigned v4u __attribute__((ext_vector_type(4)));
typedef int      v4i __attribute__((ext_vector_type(4)));

// Address-space-qualified pointers for the async builtin:
//   arg0: addrspace(1) v4i*  (global source)   [per compiler diagnostic]
//   arg1: addrspace(3) v4i*  (LDS destination)
typedef __attribute__((address_space(1))) v4i gas_v4i;
typedef __attribute__((address_space(3))) v4i las_v4i;

#if __has_builtin(__builtin_amdgcn_global_load_async_to_lds_b128)
#  define ASYNC_B128(gp, lp)                                        \
     __builtin_amdgcn_global_load_async_to_lds_b128(                \
         (gas_v4i*)(void*)(gp), (las_v4i*)(void*)(lp), 0, 0)
#else
#  define ASYNC_B128(gp, lp)                                                   \
     asm volatile("global_load_async_to_lds_b128 %0, %1, off"                  \
                  :: "v"((unsigned)(unsigned long long)(las_v4i*)(void*)(lp)), \
                     "v"((void*)(gp))                                          \
                  : "memory")
#endif

#if __has_builtin(__builtin_amdgcn_s_wait_asynccnt)
#  define WAIT_ASYNC(n) __builtin_amdgcn_s_wait_asynccnt(n)
#else
#  define WAIT_ASYNC(n) asm volatile("s_wait_asynccnt %0" :: "i"(n) : "memory")
#endif

// Monotone key: unsigned compare on key(f) == float compare on f.
__device__ __forceinline__ unsigned key_of(unsigned u) {
  return (u & 0x80000000u) ? ~u : (u | 0x80000000u);
}

__global__ __launch_bounds__(THREADS, 1)
void sparsify_topk_kernel(const float* __restrict__ x, float* __restrict__ out)
{
  const int tid = threadIdx.x;
  const v4f* row4 = (const v4f*)x + (size_t)blockIdx.x * ROW_VEC;
  v4u*       out4 = (v4u*)out     + (size_t)blockIdx.x * ROW_VEC;

  __shared__ unsigned hist[256 * NBANK];   // banked histogram
  __shared__ unsigned histsum[256];
  __shared__ unsigned bcast[2];            // {selected bin, remaining k}
  __shared__ v4f      stage[2][THREADS];   // async double buffer (32 KiB)

  const int bank = tid & (NBANK - 1);
  unsigned prefix = 0;       // selected high bits so far (right-aligned)
  unsigned kRem   = K_KEEP;  // rank to find among prefix-matching keys

  for (int round = 0; round < 4; ++round) {
    const int shift = 24 - 8 * round;

    // Zero banked histogram.
    for (int i = tid; i < 256 * NBANK; i += THREADS) hist[i] = 0;

    // Prime the async pipeline: each lane copies its own 16B slot.
    ASYNC_B128(row4 + tid, &stage[0][tid]);
    __syncthreads();  // histogram zero visible before atomics

    auto accum = [&](v4f v) {
#pragma unroll
      for (int j = 0; j < 4; ++j) {
        unsigned key = key_of(__float_as_uint(v[j]));
        // round 0: shift+8 == 32 -> 64-bit shift gives 0 == prefix(0): no filter.
        if (((unsigned long long)key >> (shift + 8)) == (unsigned long long)prefix)
          atomicAdd(&hist[((key >> shift) & 255u) * NBANK + bank], 1u);
      }
    };

    // Double-buffered async streaming: keep 1 copy in flight, wait on own wave.
    for (int t = 0; t < NTILES - 1; ++t) {
      ASYNC_B128(row4 + (t + 1) * THREADS + tid, &stage[(t + 1) & 1][tid]);
      WAIT_ASYNC(1);                 // tile t landed in LDS
      accum(stage[t & 1][tid]);
    }
    WAIT_ASYNC(0);
    accum(stage[(NTILES - 1) & 1][tid]);

    __syncthreads();

    // Reduce the 16 banks per bin.
    if (tid < 256) {
      unsigned s = 0;
#pragma unroll
      for (int j = 0; j < NBANK; ++j) s += hist[tid * NBANK + j];
      histsum[tid] = s;
    }
    __syncthreads();

    // Descending scan: find the bin holding the kRem-th largest key.
    if (tid == 0) {
      unsigned cum = 0;
      for (int b = 255; b >= 0; --b) {
        unsigned c = histsum[b];
        if (cum + c >= kRem) { bcast[0] = (unsigned)b; bcast[1] = kRem - cum; break; }
        cum += c;
      }
    }
    __syncthreads();

    prefix = (prefix << 8) | bcast[0];
    kRem   = bcast[1];
  }

  // prefix is now the exact 32-bit key of the k-th largest element.
  const unsigned thrKey = prefix;

  // Output pass: x is L2-resident; write out with NT stores so it stays hot.
  for (int t = 0; t < NTILES; ++t) {
    v4f v = row4[t * THREADS + tid];
    v4u r;
#pragma unroll
    for (int j = 0; j < 4; ++j) {   // keep x, else 0*x (preserves signed zero)
      unsigned u = __float_as_uint(v[j]);
      r[j] = (key_of(u) >= thrKey) ? u : (u & 0x80000000u);
    }
    __builtin_nontemporal_store(r, &out4[t * THREADS + tid]);
  }
}

extern "C" void kernel_launch(void* const* d_in, const int* in_sizes, int n_in,
                              void* d_out, int out_size, void* d_ws, size_t ws_size,
                              hipStream_t stream) {
  const float* x = (const float*)d_in[0];
  float* out = (float*)d_out;
  const int B = in_sizes[0] / ROW_ELEMS;  // 128
  sparsify_topk_kernel<<<dim3(B), dim3(THREADS), 0, stream>>>(x, out);
}